// GNNAutoencoder_18915035972104
// MI455X (gfx1250) — compile-verified
//
#include <hip/hip_runtime.h>

#define NN 50000
#define NE 800000

typedef __attribute__((ext_vector_type(2))) float v2f;
typedef __attribute__((ext_vector_type(8))) float v8f;

// ---------------- degree / normalization ----------------
__global__ void k_init_deg(float* deg) {
  int i = blockIdx.x * blockDim.x + threadIdx.x;
  if (i < NN) deg[i] = 1.0f;                       // self-loop contributes 1
}

__global__ void k_deg_edges(const int* __restrict__ dst, float* __restrict__ deg) {
  int e = blockIdx.x * blockDim.x + threadIdx.x;
  if (e < NE)
    __hip_atomic_fetch_add(deg + dst[e], 1.0f, __ATOMIC_RELAXED, __HIP_MEMORY_SCOPE_AGENT);
}

__global__ void k_rsqrt(float* d) {
  int i = blockIdx.x * blockDim.x + threadIdx.x;
  if (i < NN) d[i] = rsqrtf(d[i]);
}

__global__ void k_zero(float* __restrict__ p, unsigned long long n) {
  unsigned long long i = (unsigned long long)blockIdx.x * blockDim.x + threadIdx.x;
  if (i < n) p[i] = 0.0f;
}

// ---------------- dense GEMM: H[M,N] = X[M,K] * W[K,N] ----------------
// One 16x16 output tile per wave32 using V_WMMA_F32_16X16X4_F32, K stepped by 4.
// A layout (32-bit 16x4): lanes 0-15 hold K=0,1 ; lanes 16-31 hold K=2,3.
// B layout (4x16)       : mirrored (rows striped across lanes within a VGPR).
// C/D layout            : VGPR r -> M = r + 8*(lane>=16), N = lane&15.
__global__ void k_gemm_wmma(const float* __restrict__ X, const float* __restrict__ W,
                            float* __restrict__ H, int K, int N) {
  int wave = (int)((blockIdx.x * blockDim.x + threadIdx.x) >> 5);
  int lane = threadIdx.x & 31;
  int tilesN = N >> 4;
  int totTiles = (NN >> 4) * tilesN;
  if (wave >= totTiles) return;                    // wave-uniform: EXEC all-1 past here
  int tr = wave / tilesN;
  int tm = tr << 4;
  int tn = (wave - tr * tilesN) << 4;
  int half = lane >> 4;
  int l    = lane & 15;

  const float* xrow = X + (size_t)(tm + l) * K + 2 * half;
  const float* wcol = W + tn + l;
  v8f c = {};
  for (int k = 0; k < K; k += 4) {
    v2f a, b;
    a.x = xrow[k];
    a.y = xrow[k + 1];
    b.x = wcol[(size_t)(k + 2 * half) * N];
    b.y = wcol[(size_t)(k + 2 * half + 1) * N];
    c = __builtin_amdgcn_wmma_f32_16x16x4_f32(false, a, false, b, (short)0, c, false, false);
  }
  float* hrow = H + (size_t)(tm + 8 * half) * N + tn + l;
#pragma unroll
  for (int r = 0; r < 8; ++r) hrow[(size_t)r * N] = c[r];
}

// ---------------- per-edge gather * norm -> atomic scatter-add ----------------
// One thread per (edge, 4-channel group); float4 gather, 4x global_atomic_add_f32.
__global__ void k_edge_scatter(const float* __restrict__ H, const int* __restrict__ src,
                               const int* __restrict__ dst, const float* __restrict__ dinv,
                               float* __restrict__ out, int C4) {
  long long gid = (long long)blockIdx.x * blockDim.x + threadIdx.x;
  int e = (int)(gid / C4);
  if (e >= NE) return;
  int c = (int)(gid - (long long)e * C4) << 2;
  int s = src[e], d = dst[e];
  if ((threadIdx.x & 63) == 0 && e + 256 < NE) {   // stream edge indices ahead (global_prefetch_b8)
    __builtin_prefetch(src + e + 256, 0, 0);
    __builtin_prefetch(dst + e + 256, 0, 0);
  }
  float nrm = dinv[s] * dinv[d];
  int C = C4 << 2;
  const float4 v = *(const float4*)(H + (size_t)s * C + c);
  float* op = out + (size_t)d * C + c;
  __hip_atomic_fetch_add(op + 0, v.x * nrm, __ATOMIC_RELAXED, __HIP_MEMORY_SCOPE_AGENT);
  __hip_atomic_fetch_add(op + 1, v.y * nrm, __ATOMIC_RELAXED, __HIP_MEMORY_SCOPE_AGENT);
  __hip_atomic_fetch_add(op + 2, v.z * nrm, __ATOMIC_RELAXED, __HIP_MEMORY_SCOPE_AGENT);
  __hip_atomic_fetch_add(op + 3, v.w * nrm, __ATOMIC_RELAXED, __HIP_MEMORY_SCOPE_AGENT);
}

// ---------------- self-loop + bias + optional ReLU ----------------
__global__ void k_finalize(float* __restrict__ out, const float* __restrict__ H,
                           const float* __restrict__ dinv, const float* __restrict__ bias,
                           int C, int relu) {
  long long gid = (long long)blockIdx.x * blockDim.x + threadIdx.x;
  if (gid >= (long long)NN * C) return;
  int i = (int)(gid / C);
  int c = (int)(gid - (long long)i * C);
  float di = dinv[i];
  float v = out[gid] + H[gid] * (di * di) + bias[c];
  out[gid] = relu ? fmaxf(v, 0.0f) : v;
}

static inline unsigned nblk(long long n, int b) { return (unsigned)((n + b - 1) / b); }

extern "C" void kernel_launch(void* const* d_in, const int* in_sizes, int n_in,
                              void* d_out, int out_size, void* d_ws, size_t ws_size,
                              hipStream_t stream) {
  (void)in_sizes; (void)n_in; (void)out_size; (void)ws_size;
  const float* x   = (const float*)d_in[0];
  const int*   ei  = (const int*)d_in[1];
  const float* We1 = (const float*)d_in[2]; const float* be1 = (const float*)d_in[3];
  const float* We2 = (const float*)d_in[4]; const float* be2 = (const float*)d_in[5];
  const float* Wd1 = (const float*)d_in[6]; const float* bd1 = (const float*)d_in[7];
  const float* Wd2 = (const float*)d_in[8]; const float* bd2 = (const float*)d_in[9];
  const int* src = ei;
  const int* dst = ei + NE;

  char*  ws   = (char*)d_ws;
  float* dinv = (float*)ws;                                  // NN floats
  size_t off  = ((size_t)NN * 4 + 255) & ~(size_t)255;
  float* G    = (float*)(ws + off);                          // NN x 128 (gemm out)
  float* P    = G + (size_t)NN * 128;                        // NN x 128 (layer out)

  // degree with self-loops -> dinv = rsqrt(deg), in place
  k_init_deg <<<nblk(NN, 256), 256, 0, stream>>>(dinv);
  k_deg_edges<<<nblk(NE, 256), 256, 0, stream>>>(dst, dinv);
  k_rsqrt    <<<nblk(NN, 256), 256, 0, stream>>>(dinv);

  struct Layer { const float* in; const float* W; const float* b; int K; int C; int relu; float* S; };
  Layer layers[4] = {
    { x, We1, be1, 128,  64, 1, P },
    { P, We2, be2,  64,  32, 0, P },
    { P, Wd1, bd1,  32,  64, 1, P },
    { P, Wd2, bd2,  64, 128, 0, (float*)d_out },
  };

  for (int li = 0; li < 4; ++li) {
    const Layer& L = layers[li];
    long long tot   = (long long)NN * L.C;
    long long tiles = (long long)(NN / 16) * (L.C / 16);
    k_gemm_wmma   <<<nblk(tiles * 32, 256), 256, 0, stream>>>(L.in, L.W, G, L.K, L.C);
    k_zero        <<<nblk(tot, 256),        256, 0, stream>>>(L.S, (unsigned long long)tot);
    k_edge_scatter<<<nblk((long long)NE * (L.C / 4), 256), 256, 0, stream>>>(G, src, dst, dinv, L.S, L.C / 4);
    k_finalize    <<<nblk(tot, 256),        256, 0, stream>>>(L.S, G, dinv, L.b, L.C, L.relu);
  }
}